// SemanticGroupProcessor_1125281431597
// MI455X (gfx1250) — compile-verified
//
#include <hip/hip_runtime.h>
#include <hip/hip_bf16.h>

// ST-GCN (3 layers) on MI455X / gfx1250 using fp32 WMMA (V_WMMA_F32_16X16X4_F32).
//
// Algebra:
//  * K-partition fold -> sg_w collapses to W_eff[cout,cin] (summed over K slabs).
//  * adj (V-mixing) commutes with the 1x1 conv -> fused into GEMM1 B-loads.
//  * Eval-mode BNs folded into weights/biases by a tiny device prep kernel.
//  * Temporal (3,1) conv + residual 1x1 conv fused into one GEMM with
//    K = 3*cout (t-shifted Z reads) + cin_pad (residual X reads).
//
// Internal activation layout: [channel][position], position p = (b*512 + t)*6 + v,
// P = 64*512*6 = 196608. Layer-1 input read directly from (B,3,T,V) layout.
//
// Workspace (floats): Z[256P] | YB[256P] | YD[128P] | folded weights (~0.34M)
//   ~505 MB total. Guarded against short ws_size.

typedef float v2f __attribute__((ext_vector_type(2)));
typedef float v8f __attribute__((ext_vector_type(8)));

#define P_TOT 196608u   // 64*512*6
#define TV    3072u     // 512*6

// ---------------------------------------------------------------------------
// Fold BN / partition-sum into GEMM-ready weight matrices. One block, 256 thr.
// A1: cout x cin_pad   (spatial GEMM weights, padded cols zeroed)
// A2: cout x (3*cout + cin_pad)  [tcw(dt=-1) | tcw(0) | tcw(+1) | res] * bn_scale
// ---------------------------------------------------------------------------
__global__ void prep_kernel(const float* __restrict__ sg_w, const float* __restrict__ sg_b,
                            const float* __restrict__ tc_w, const float* __restrict__ tc_b,
                            const float* __restrict__ res_w, const float* __restrict__ res_b,
                            const float* __restrict__ res_g, const float* __restrict__ res_beta,
                            const float* __restrict__ res_m, const float* __restrict__ res_v,
                            const float* __restrict__ bn_g,  const float* __restrict__ bn_beta,
                            const float* __restrict__ bn_m,  const float* __restrict__ bn_v,
                            float* __restrict__ A1, float* __restrict__ b1,
                            float* __restrict__ A2, float* __restrict__ b2,
                            int cout, int cin, int cin_pad)
{
    int o = threadIdx.x;
    if (o >= cout) return;
    const float EPS = 1e-5f;
    float sr = res_g[o] * rsqrtf(res_v[o] + EPS);   // residual BN scale
    float sm = bn_g[o]  * rsqrtf(bn_v[o]  + EPS);   // main BN scale

    // spatial GEMM: sum the K=3 partition slabs
    float bb = 0.f;
    for (int k = 0; k < 3; ++k) bb += sg_b[k * cout + o];
    b1[o] = bb;
    for (int c = 0; c < cin_pad; ++c) {
        float w = 0.f;
        if (c < cin)
            for (int k = 0; k < 3; ++k) w += sg_w[(k * cout + o) * cin + c];
        A1[o * cin_pad + c] = w;
    }

    int Kt = 3 * cout + cin_pad;
    // temporal conv weights, scaled by main-BN scale. tc_w OIHW (cout,cout,3,1); h=0 -> dt=-1
    for (int g = 0; g < 3; ++g)
        for (int i = 0; i < cout; ++i)
            A2[o * Kt + g * cout + i] = sm * tc_w[(o * cout + i) * 3 + g];
    // residual 1x1, scaled by both BN scales
    for (int c = 0; c < cin_pad; ++c)
        A2[o * Kt + 3 * cout + c] = (c < cin) ? sm * sr * res_w[o * cin + c] : 0.f;
    // combined bias: relu( sm*(TC + tc_b + sr*(res + res_b - res_m) + res_beta - bn_m) + bn_beta )
    b2[o] = sm * (tc_b[o] + sr * (res_b[o] - res_m[o]) + res_beta[o] - bn_m[o]) + bn_beta[o];
}

// ---------------------------------------------------------------------------
// WMMA GEMM: Out[M][P] = op(A[M][Kt] * B[Kt][P] + bias), 4 waves/block,
// each wave = one 16-row M tile x four 16-col N tiles.
//   AGG  : B rows are channels of X with adj aggregation fused (stage-1 GEMM)
//   !AGG : first tlen rows are t-shifted Z, tail rows are residual X (stage-2)
//   L1X  : X is the external (B,3,T,V) input; else internal [C][P] layout
//   RELU : compile-time epilogue activation (stage-2 only)
// ---------------------------------------------------------------------------
template<bool AGG, bool L1X, bool RELU>
__global__ __launch_bounds__(128)
void stgcn_gemm(const float* __restrict__ A, const float* __restrict__ bias,
                const float* __restrict__ Zsrc, const float* __restrict__ Xsrc,
                const float* __restrict__ adjp, float* __restrict__ Out,
                int M, int Kt, int tlen, int cin, int cimask, int cishift)
{
    const unsigned CS = L1X ? TV : P_TOT;        // channel stride of X
    unsigned lane = threadIdx.x & 31u;
    unsigned wv   = threadIdx.x >> 5;
    unsigned n    = lane & 15u;                  // N index inside tile
    unsigned kh   = lane >> 4;                   // lane-half -> K pair / row-half
    unsigned m0   = (blockIdx.y * 4u + wv) * 16u;
    unsigned p0   = blockIdx.x * 64u;

    unsigned pj[4], xoff[4], aggb[4];
    int vdm[4], vdp[4];
    float ar[4][6];
#pragma unroll
    for (int j = 0; j < 4; ++j) {
        unsigned p = p0 + (unsigned)j * 16u + n;
        pj[j] = p;
        unsigned b = p / TV;
        unsigned r = p - b * TV;
        unsigned t = r / 6u;
        unsigned u = r - t * 6u;
        if (AGG) {
            aggb[j] = L1X ? (b * 9216u + r - u) : (p - u);
#pragma unroll
            for (int v = 0; v < 6; ++v) ar[j][v] = adjp[u * 6u + v];
        } else {
            vdm[j]  = (t > 0u);
            vdp[j]  = (t < 511u);
            xoff[j] = L1X ? (b * 9216u + r) : p;
        }
    }

    auto loadB = [&](int k, int j) -> float {
        if constexpr (AGG) {
            int ok = (k < cin);
            unsigned c = ok ? (unsigned)k : 0u;
            const float* xb = Xsrc + aggb[j] + c * CS;
            float s = ar[j][0] * xb[0] + ar[j][1] * xb[1] + ar[j][2] * xb[2]
                    + ar[j][3] * xb[3] + ar[j][4] * xb[4] + ar[j][5] * xb[5];
            return ok ? s : 0.f;
        } else {
            if (k < tlen) {                       // temporal part over Z, dt = g-1
                unsigned i = (unsigned)k & (unsigned)cimask;
                int g  = k >> cishift;
                int ok = (g == 1) ? 1 : (g == 0 ? vdm[j] : vdp[j]);
                int d  = ok ? (g - 1) * 6 : 0;    // clamped address, masked value
                float v = Zsrc[(int)(i * P_TOT + pj[j]) + d];
                return ok ? v : 0.f;
            } else {                              // residual part over X
                int c  = k - tlen;
                int ok = (c < cin);
                unsigned ce = ok ? (unsigned)c : 0u;
                float v = Xsrc[xoff[j] + ce * CS];
                return ok ? v : 0.f;
            }
        }
    };

    // init accumulators with bias (C/D layout: vreg r -> row r + kh*8)
    v8f bf;
#pragma unroll
    for (int r = 0; r < 8; ++r) bf[r] = bias[m0 + kh * 8u + (unsigned)r];
    v8f acc[4];
#pragma unroll
    for (int j = 0; j < 4; ++j) acc[j] = bf;

    unsigned arow = (m0 + n) * (unsigned)Kt;      // A layout: lane m = n, k = kh*2+{0,1}
    for (int kc = 0; kc < Kt; kc += 4) {
        int k0 = kc + (int)(kh * 2u);
        v2f av;
        av.x = A[arow + (unsigned)k0];
        av.y = A[arow + (unsigned)k0 + 1u];
#pragma unroll
        for (int j = 0; j < 4; ++j) {
            v2f bv;
            bv.x = loadB(k0,     j);
            bv.y = loadB(k0 + 1, j);
            acc[j] = __builtin_amdgcn_wmma_f32_16x16x4_f32(
                false, av, false, bv, (short)0, acc[j], false, false);
        }
    }

#pragma unroll
    for (int j = 0; j < 4; ++j)
#pragma unroll
        for (int r = 0; r < 8; ++r) {
            float v = acc[j][r];
            if constexpr (RELU) v = fmaxf(v, 0.f);
            Out[(size_t)(m0 + kh * 8u + (unsigned)r) * P_TOT + p0 + (unsigned)j * 16u + n] = v;
        }
}

// ---------------------------------------------------------------------------
// AdaptiveAvgPool over (T,V): one block per (b,o); 3072 contiguous floats.
// ---------------------------------------------------------------------------
__global__ void mean_kernel(const float* __restrict__ Y, float* __restrict__ out)
{
    unsigned b = blockIdx.x >> 8;
    unsigned o = blockIdx.x & 255u;
    const float* src = Y + (size_t)o * P_TOT + b * TV;
    float s = 0.f;
    for (unsigned i = threadIdx.x; i < TV; i += 256u) s += src[i];
    __shared__ float red[256];
    red[threadIdx.x] = s;
    __syncthreads();
    for (int st = 128; st > 0; st >>= 1) {
        if ((int)threadIdx.x < st) red[threadIdx.x] += red[threadIdx.x + st];
        __syncthreads();
    }
    if (threadIdx.x == 0) out[blockIdx.x] = red[0] * (1.f / 3072.f);
}

// ---------------------------------------------------------------------------
extern "C" void kernel_launch(void* const* d_in, const int* in_sizes, int n_in,
                              void* d_out, int out_size, void* d_ws, size_t ws_size,
                              hipStream_t stream)
{
    (void)in_sizes; (void)out_size;
    const size_t P = P_TOT;
    float* ws = (float*)d_ws;
    float* Z  = ws;               // 256P : pre-activation spatial output
    float* YB = ws + 256 * P;     // 256P : Y1 (=X2), later Y3
    float* YD = ws + 512 * P;     // 128P : Y2 (=X3)
    float* Wp = ws + 640 * P;     // folded weights

    const int cins[3]  = {3, 64, 128};
    const int couts[3] = {64, 128, 256};
    const int cinp[3]  = {4, 64, 128};   // cin padded to multiple of 4

    float *A1p[3], *b1p[3], *A2p[3], *b2p[3];
    size_t woff = 0;
    for (int l = 0; l < 3; ++l) {
        int Kt2 = 3 * couts[l] + cinp[l];
        A1p[l] = Wp + woff; woff += (size_t)couts[l] * cinp[l];
        b1p[l] = Wp + woff; woff += couts[l];
        A2p[l] = Wp + woff; woff += (size_t)couts[l] * Kt2;
        b2p[l] = Wp + woff; woff += couts[l];
    }
    size_t need = (640 * P + woff) * sizeof(float);
    if (ws_size < need || n_in < 44) return;   // avoid OOB if workspace too small

    const float* x0  = (const float*)d_in[0];
    const float* adj = (const float*)d_in[1];

    for (int l = 0; l < 3; ++l) {
        const float* const* pp = (const float* const*)(d_in + 2 + l * 14);
        prep_kernel<<<1, 256, 0, stream>>>(
            pp[0], pp[1], pp[2], pp[3], pp[4], pp[5], pp[6], pp[7],
            pp[8], pp[9], pp[10], pp[11], pp[12], pp[13],
            A1p[l], b1p[l], A2p[l], b2p[l], couts[l], cins[l], cinp[l]);
    }

    dim3 blk(128);
    unsigned gx = (unsigned)(P / 64);   // 3072 N-supertiles

    // Layer 1 (3 -> 64), X in external (B,3,T,V) layout
    stgcn_gemm<true,  true,  false><<<dim3(gx, 1), blk, 0, stream>>>(
        A1p[0], b1p[0], nullptr, x0, adj, Z,  64, 4,   0,   3, 0,   0);
    stgcn_gemm<false, true,  true ><<<dim3(gx, 1), blk, 0, stream>>>(
        A2p[0], b2p[0], Z, x0, adj, YB,       64, 196, 192, 3, 63,  6);

    // Layer 2 (64 -> 128)
    stgcn_gemm<true,  false, false><<<dim3(gx, 2), blk, 0, stream>>>(
        A1p[1], b1p[1], nullptr, YB, adj, Z,  128, 64,  0,   64, 0,   0);
    stgcn_gemm<false, false, true ><<<dim3(gx, 2), blk, 0, stream>>>(
        A2p[1], b2p[1], Z, YB, adj, YD,       128, 448, 384, 64, 127, 7);

    // Layer 3 (128 -> 256)
    stgcn_gemm<true,  false, false><<<dim3(gx, 4), blk, 0, stream>>>(
        A1p[2], b1p[2], nullptr, YD, adj, Z,  256, 128, 0,   128, 0,   0);
    stgcn_gemm<false, false, true ><<<dim3(gx, 4), blk, 0, stream>>>(
        A2p[2], b2p[2], Z, YD, adj, YB,       256, 896, 768, 128, 255, 8);

    // Global average pool -> (B, 256)
    mean_kernel<<<64 * 256, 256, 0, stream>>>(YB, (float*)d_out);
}